// Pi0_44959717654950
// MI455X (gfx1250) — compile-verified
//
#include <hip/hip_runtime.h>
#include <hip/hip_bf16.h>
#include <math.h>

typedef __attribute__((ext_vector_type(2))) float v2f;
typedef __attribute__((ext_vector_type(4))) float v4f;
typedef __attribute__((ext_vector_type(8))) float v8f;

#define WIDTH   1024
#define BDIM    32
#define HDIM    50
#define SDIM    32
#define PDIM    1997
#define NTOK    2048              // P + 1 + H
#define MROWS   (BDIM * HDIM)     // 1600
#define SUFFIX_ELEMS (BDIM * (1 + HDIM) * WIDTH)  // 1671168

// ---------------------------------------------------------------------------
// Time embedding: att_in[row, 1024+j] = sin(ang), att_in[row, 1536+j] = cos(ang)
// ang = t * 2*pi / (MIN * (MAX/MIN)^(j/511))
// ---------------------------------------------------------------------------
__global__ void time_embed_kernel(const float* __restrict__ timestep,
                                  float* __restrict__ att_in) {
    const int row = blockIdx.x;          // 0..1599
    const int j   = threadIdx.x;         // 0..511
    const int b   = row / HDIM;
    const float t = timestep[b];
    const float frac = (float)j * (1.0f / 511.0f);
    // period = 0.004 * 1000^frac  ->  1/period = 250 * exp(-frac*ln(1000))
    const float inv_p = 250.0f * __expf(-frac * 6.9077552789821f);
    const float ang = t * 6.28318530717958647f * inv_p;
    float* dst = att_in + (size_t)row * (2 * WIDTH) + WIDTH;
    dst[j]       = __sinf(ang);
    dst[512 + j] = __cosf(ang);
}

// ---------------------------------------------------------------------------
// State token: out[b*51*1024 + n] = numeric_state[b,:] @ W_state[:,n] + b_state[n]
// ---------------------------------------------------------------------------
__global__ void state_token_kernel(const float* __restrict__ ns,
                                   const float* __restrict__ W,
                                   const float* __restrict__ bvec,
                                   float* __restrict__ out) {
    const int idx = blockIdx.x * blockDim.x + threadIdx.x;  // b*1024 + n
    const int b = idx >> 10;
    const int n = idx & 1023;
    float s = bvec[n];
    #pragma unroll
    for (int k = 0; k < SDIM; ++k)
        s += ns[b * SDIM + k] * W[k * WIDTH + n];
    out[(size_t)(b * (1 + HDIM)) * WIDTH + n] = s;
}

// ---------------------------------------------------------------------------
// fp32 WMMA GEMM: C = A[MxK] * B[KxN] + bias, optional swish / scatter.
//   mode 0: C[m*ldc + n]           (bias)
//   mode 1: C[m*ldc + n]           (bias + swish)
//   mode 2: C[(b*51 + 1 + h)*1024 + n], m = b*50 + h   (bias, suffix scatter)
// Block = 256 threads = 8 waves, block tile 32(M) x 128(N), BK = 16.
// Each wave: one 16x16 M-tile, two 16x16 N-tiles (A-fragment reuse).
// B tile staged via GLOBAL_LOAD_ASYNC_TO_LDS_B128 (ASYNCcnt path, no VGPR
// round-trip); A tile staged via DS stores (needs x17-padded scatter).
// ---------------------------------------------------------------------------
__global__ __launch_bounds__(256)
void gemm_wmma_f32(const float* __restrict__ A, const float* __restrict__ B,
                   const float* __restrict__ bias, float* __restrict__ C,
                   int M, int K, int N, int ldc, int mode) {
    __shared__ float As[32 * 17];    // padded: row stride 17 floats
    __shared__ float Bs[16 * 132];   // padded: row stride 132 floats

    const int tid  = threadIdx.x;
    const int lane = tid & 31;
    const int wid  = tid >> 5;
    const int wm   = wid >> 2;               // 0..1  -> M sub-tile
    const int wn   = wid & 3;                // 0..3  -> N sub-group (32 wide)
    const int m0   = blockIdx.y * 32;
    const int n0   = blockIdx.x * 128;

    const int lm      = lane & 15;           // row-in-tile / col-in-tile
    const int kb_half = (lane >> 4) * 2;     // 0 or 2 (K half per ISA layout)

    v8f acc0 = {};
    v8f acc1 = {};

    // per-thread staging coordinates (fixed across K loop)
    const int bk = (tid * 8) >> 7;           // B: k row 0..15
    const int bn = (tid * 8) & 127;          // B: n col (multiple of 8)
    const unsigned ldsb =
        (unsigned)(uintptr_t)(&Bs[bk * 132 + bn]);   // LDS byte address (low 32b)

    for (int k0 = 0; k0 < K; k0 += 16) {
        // ---- stage A: 32x16 (512 floats, 2 per thread) via DS stores ----
        {
            const int idx = tid * 2;
            const int m = idx >> 4;
            const int k = idx & 15;
            const float2 a2 = *(const float2*)(A + (size_t)(m0 + m) * K + k0 + k);
            As[m * 17 + k]     = a2.x;
            As[m * 17 + k + 1] = a2.y;
        }
        // ---- stage B: 16x128 via async global->LDS b128 (2 per thread) ----
        {
            const float* src = B + (size_t)(k0 + bk) * N + n0 + bn;
            asm volatile("global_load_async_to_lds_b128 %0, %1, off"
                         :: "v"(ldsb),
                            "v"((unsigned long long)(uintptr_t)src)
                         : "memory");
            asm volatile("global_load_async_to_lds_b128 %0, %1, off"
                         :: "v"(ldsb + 16u),
                            "v"((unsigned long long)(uintptr_t)(src + 4))
                         : "memory");
            if (k0 + 16 < K)   // prefetch next B tile -> global_prefetch
                __builtin_prefetch(src + (size_t)16 * N, 0, 1);
        }
        asm volatile("s_wait_asynccnt 0x0" ::: "memory");
        __syncthreads();

        #pragma unroll
        for (int kk = 0; kk < 16; kk += 4) {
            // A frag (16x4): lane l -> m = l&15, k = kk + 2*(l>>4) + {0,1}
            v2f af;
            af.x = As[(wm * 16 + lm) * 17 + kk + kb_half];
            af.y = As[(wm * 16 + lm) * 17 + kk + kb_half + 1];
            // B frags (4x16): lane l -> n = l&15, k = kk + 2*(l>>4) + {0,1}
            const int nb = wn * 32;
            v2f bf0, bf1;
            bf0.x = Bs[(kk + kb_half)     * 132 + nb + lm];
            bf0.y = Bs[(kk + kb_half + 1) * 132 + nb + lm];
            bf1.x = Bs[(kk + kb_half)     * 132 + nb + 16 + lm];
            bf1.y = Bs[(kk + kb_half + 1) * 132 + nb + 16 + lm];
            acc0 = __builtin_amdgcn_wmma_f32_16x16x4_f32(
                       false, af, false, bf0, (short)0, acc0, false, false);
            acc1 = __builtin_amdgcn_wmma_f32_16x16x4_f32(
                       false, af, false, bf1, (short)0, acc1, false, false);
        }
        __syncthreads();
    }

    // ---- epilogue: D layout lane(0-15)->M=v, lane(16-31)->M=v+8, N=lane&15 ----
    const int nA = n0 + wn * 32 + lm;
    const int nB = nA + 16;
    const float biasA = bias[nA];
    const float biasB = bias[nB];
    const int mbase = m0 + wm * 16 + (lane >> 4) * 8;

    #pragma unroll
    for (int v = 0; v < 8; ++v) {
        const int m = mbase + v;
        float x0 = acc0[v] + biasA;
        float x1 = acc1[v] + biasB;
        if (mode == 1) {                       // swish: x * sigmoid(x)
            x0 = x0 / (1.0f + __expf(-x0));
            x1 = x1 / (1.0f + __expf(-x1));
        }
        if (mode == 2) {                       // scatter into suffix_tokens
            const int b = m / HDIM, h = m % HDIM;
            const size_t ro = (size_t)(b * (1 + HDIM) + 1 + h) * WIDTH;
            C[ro + nA] = x0;
            C[ro + nB] = x1;
        } else {
            C[(size_t)m * ldc + nA] = x0;
            C[(size_t)m * ldc + nB] = x1;
        }
    }
}

// ---------------------------------------------------------------------------
// Mask prep: cumsum of ar mask (same for all b) and per-batch validity.
// ---------------------------------------------------------------------------
__global__ void mask_prep_kernel(const unsigned char* __restrict__ prefix_in_mask,
                                 const unsigned char* __restrict__ prefix_ar,
                                 int* __restrict__ c,
                                 unsigned char* __restrict__ valid) {
    if (threadIdx.x == 0) {
        int s = 0;
        for (int j = 0; j < NTOK; ++j) {
            int a;
            if (j < PDIM)                      a = prefix_ar[j] ? 1 : 0;
            else if (j == PDIM || j == PDIM+1) a = 1;   // state token + 1st action
            else                               a = 0;
            s += a;
            c[j] = s;
        }
    }
    for (int idx = threadIdx.x; idx < BDIM * NTOK; idx += blockDim.x) {
        const int b = idx >> 11;
        const int j = idx & (NTOK - 1);
        valid[idx] = (j < PDIM) ? (prefix_in_mask[b * PDIM + j] ? 1 : 0) : 1;
    }
}

// ---------------------------------------------------------------------------
// Mask fill (bandwidth-bound): one block per (b, i) row, 32 B / thread as
// non-temporal b128 stores (537 MB streaming output; keep it out of L2).
// mask[b,i,j] = (c[j] <= c[i]) && valid[b,i] && valid[b,j]
// ---------------------------------------------------------------------------
__global__ __launch_bounds__(256)
void mask_fill_kernel(const int* __restrict__ c,
                      const unsigned char* __restrict__ valid,
                      float* __restrict__ mask) {
    const int bi = blockIdx.x;               // b*2048 + i
    const int b  = bi >> 11;
    const int ci = c[bi & (NTOK - 1)];
    const int vi = valid[bi];
    float* row = mask + (size_t)bi * NTOK;
    const unsigned char* vb = valid + ((size_t)b << 11);
    const int j0 = threadIdx.x * 8;

    float vals[8];
    #pragma unroll
    for (int u = 0; u < 8; ++u) {
        const int j = j0 + u;
        vals[u] = (vi && vb[j] && (c[j] <= ci)) ? 1.0f : 0.0f;
    }
    const v4f lo = {vals[0], vals[1], vals[2], vals[3]};
    const v4f hi = {vals[4], vals[5], vals[6], vals[7]};
    __builtin_nontemporal_store(lo, (v4f*)(row + j0));
    __builtin_nontemporal_store(hi, (v4f*)(row + j0 + 4));
}

// ---------------------------------------------------------------------------
extern "C" void kernel_launch(void* const* d_in, const int* in_sizes, int n_in,
                              void* d_out, int out_size, void* d_ws, size_t ws_size,
                              hipStream_t stream) {
    const unsigned char* prefix_input_mask = (const unsigned char*)d_in[0];
    const unsigned char* prefix_ar_mask    = (const unsigned char*)d_in[1];
    const float* timestep      = (const float*)d_in[2];
    const float* numeric_state = (const float*)d_in[3];
    const float* noisy_actions = (const float*)d_in[4];
    const float* W_state   = (const float*)d_in[5];
    const float* b_state   = (const float*)d_in[6];
    const float* W_act     = (const float*)d_in[7];
    const float* b_act     = (const float*)d_in[8];
    const float* W_mlp_in  = (const float*)d_in[9];
    const float* b_mlp_in  = (const float*)d_in[10];
    const float* W_mlp_out = (const float*)d_in[11];
    const float* b_mlp_out = (const float*)d_in[12];

    float* out = (float*)d_out;

    // workspace layout
    char* ws = (char*)d_ws;
    float* att_in = (float*)ws;                                   // 1600 x 2048
    float* hmid   = (float*)(ws + (size_t)MROWS * 2 * WIDTH * 4); // 1600 x 1024
    int*   c      = (int*)  (ws + (size_t)MROWS * 3 * WIDTH * 4);
    unsigned char* valid = (unsigned char*)(c + NTOK);

    // 1) time embedding -> att_in[:, 1024:2048]
    time_embed_kernel<<<MROWS, 512, 0, stream>>>(timestep, att_in);

    // 2) action tokens: [1600,32]@[32,1024] + b_act -> att_in[:, 0:1024]
    gemm_wmma_f32<<<dim3(WIDTH / 128, MROWS / 32), 256, 0, stream>>>(
        noisy_actions, W_act, b_act, att_in, MROWS, SDIM, WIDTH, 2 * WIDTH, 0);

    // 3) state token -> out[b, 0, :]
    state_token_kernel<<<(BDIM * WIDTH) / 256, 256, 0, stream>>>(
        numeric_state, W_state, b_state, out);

    // 4) MLP in + swish: [1600,2048]@[2048,1024] -> hmid
    gemm_wmma_f32<<<dim3(WIDTH / 128, MROWS / 32), 256, 0, stream>>>(
        att_in, W_mlp_in, b_mlp_in, hmid, MROWS, 2 * WIDTH, WIDTH, WIDTH, 1);

    // 5) MLP out: [1600,1024]@[1024,1024] -> out[b, 1+h, :]
    gemm_wmma_f32<<<dim3(WIDTH / 128, MROWS / 32), 256, 0, stream>>>(
        hmid, W_mlp_out, b_mlp_out, out, MROWS, WIDTH, WIDTH, WIDTH, 2);

    // 6) attention mask
    mask_prep_kernel<<<1, 256, 0, stream>>>(prefix_input_mask, prefix_ar_mask, c, valid);
    mask_fill_kernel<<<BDIM * NTOK, 256, 0, stream>>>(c, valid, out + SUFFIX_ELEMS);
}